// Sober_14903536517471
// MI455X (gfx1250) — compile-verified
//
#include <hip/hip_runtime.h>
#include <hip/hip_bf16.h>

// 3D Sobel magnitude: out = sqrt(gx^2 + gy^2 + 1e-6), depthwise 3x3x3 over
// [N=2, C=32, D=96, H=96, W=96] f32, zero padding.
// Memory-bound: stage halo tiles in LDS via CDNA5 async global->LDS loads,
// then compute with separable factorization and per-plane partial reuse:
//   KX = s_d (x) s_h (x) g_w,  KY = s_d (x) g_h (x) s_w,  s=[1,2,1], g=[1,0,-1]
//   a[z] = H-smooth(W-deriv)(plane z); b[z] = H-deriv(W-smooth)(plane z)
//   gx(dz) = a[dz] + 2a[dz+1] + a[dz+2];  gy(dz) = b[dz] + 2b[dz+1] + b[dz+2]

#define DIM 96          // D == H == W == 96
#define TW 32           // tile width  (== lanes per wave)
#define TH 16           // tile height
#define TD 8            // tile depth (outputs per thread, one D-column)
#define HW (TW + 2)     // halo dims: 34
#define HH (TH + 2)     // 18
#define HD (TD + 2)     // 10
#define HALO_ELEMS (HD * HH * HW)   // 10*18*34 = 6120 floats (24480 B LDS)
#define NTHREADS (TW * TH)          // 512 threads = 16 waves

typedef __attribute__((address_space(3))) float lds_f32;

__device__ __forceinline__ void async_load_f32_to_lds(lds_f32* lds_dst,
                                                      const float* gsrc) {
    // GV mode: VDST = LDS byte offset (1 VGPR), VADDR = 64-bit global addr.
    asm volatile("global_load_async_to_lds_b32 %0, %1, off"
                 :
                 : "v"(lds_dst), "v"(gsrc)
                 : "memory");
}

__device__ __forceinline__ void wait_async_zero() {
#if __has_builtin(__builtin_amdgcn_s_wait_asynccnt)
    __builtin_amdgcn_s_wait_asynccnt(0);
#else
    asm volatile("s_wait_asynccnt 0" ::: "memory");
#endif
}

__global__ __launch_bounds__(NTHREADS)
void sobel3d_kernel(const float* __restrict__ x, float* __restrict__ out) {
    __shared__ float tile[HALO_ELEMS];

    const int bw = blockIdx.x;            // 0..2   (96/32)
    const int bh = blockIdx.y;            // 0..5   (96/16)
    const int bz = blockIdx.z;            // (96/8) * B
    const int bd = bz % (DIM / TD);
    const int b  = bz / (DIM / TD);       // fused N*C index

    const int w0 = bw * TW;
    const int h0 = bh * TH;
    const int d0 = bd * TD;
    const size_t volBase = (size_t)b * (DIM * DIM * DIM);

    const int t = threadIdx.x;

    // ---- stage halo tile into LDS (async), zeros for out-of-volume ----
    for (int e = t; e < HALO_ELEMS; e += NTHREADS) {
        const int dz = e / (HH * HW);
        const int r  = e % (HH * HW);
        const int dy = r / HW;
        const int dx = r % HW;
        const int gz = d0 - 1 + dz;
        const int gy = h0 - 1 + dy;
        const int gw = w0 - 1 + dx;
        const bool inRange = ((unsigned)gz < (unsigned)DIM) &&
                             ((unsigned)gy < (unsigned)DIM) &&
                             ((unsigned)gw < (unsigned)DIM);
        if (inRange) {
            const float* gp = x + volBase + ((size_t)gz * DIM + gy) * DIM + gw;
            async_load_f32_to_lds((lds_f32*)(tile + e), gp);
        } else {
            tile[e] = 0.0f;   // ds_store; fenced by __syncthreads()
        }
    }
    wait_async_zero();
    __syncthreads();

    // ---- compute: lanes along W, ty along H, one D-column per thread ----
    const int tx = t & (TW - 1);
    const int ty = t >> 5;                // 0..15

    float a[HD], bb[HD];                  // per-plane partials
#pragma unroll
    for (int z = 0; z < HD; ++z) {
        const float* p = &tile[(z * HH + ty) * HW + tx];
        const float v00 = p[0],      v01 = p[1],          v02 = p[2];
        const float v10 = p[HW],     v12 = p[HW + 2];
        const float v20 = p[2 * HW], v21 = p[2 * HW + 1], v22 = p[2 * HW + 2];
        const float rs0 = v00 + 2.0f * v01 + v02;   // W-smooth, row 0
        const float rs2 = v20 + 2.0f * v21 + v22;   // W-smooth, row 2
        const float rd0 = v00 - v02;                // W-deriv rows
        const float rd1 = v10 - v12;
        const float rd2 = v20 - v22;
        a[z]  = rd0 + 2.0f * rd1 + rd2;             // H-smooth of W-deriv
        bb[z] = rs0 - rs2;                          // H-deriv  of W-smooth
    }

    const size_t outBase = volBase + ((size_t)d0 * DIM + (h0 + ty)) * DIM + (w0 + tx);
#pragma unroll
    for (int dz = 0; dz < TD; ++dz) {
        const float gxv = a[dz]  + 2.0f * a[dz + 1]  + a[dz + 2];
        const float gyv = bb[dz] + 2.0f * bb[dz + 1] + bb[dz + 2];
        out[outBase + (size_t)dz * (DIM * DIM)] =
            __builtin_sqrtf(gxv * gxv + gyv * gyv + 1e-6f);
    }
}

extern "C" void kernel_launch(void* const* d_in, const int* in_sizes, int n_in,
                              void* d_out, int out_size, void* d_ws, size_t ws_size,
                              hipStream_t stream) {
    (void)n_in; (void)out_size; (void)d_ws; (void)ws_size;
    const float* x = (const float*)d_in[0];
    float* out = (float*)d_out;

    const int B = in_sizes[0] / (DIM * DIM * DIM);   // N*C = 64
    dim3 grid(DIM / TW, DIM / TH, (DIM / TD) * B);   // 3 x 6 x 768
    sobel3d_kernel<<<grid, NTHREADS, 0, stream>>>(x, out);
}